// MHR_76965813945133
// MI455X (gfx1250) — compile-verified
//
#include <hip/hip_runtime.h>

typedef __attribute__((ext_vector_type(2))) float v2f;
typedef __attribute__((ext_vector_type(8))) float v8f;

#define D_MODEL   1024
#define NUM_HEADS 16
#define HEAD_DIM  64
#define SEQ       2048
#define BATCH     2
#define NEGMASK   (-10000.0f)
#define KSCALE    0.125f   /* 1/sqrt(64) */

// ---- optional gfx1250 async global->LDS path (compile-guarded probe) ------
#if defined(__has_builtin)
#  if __has_builtin(__builtin_amdgcn_global_load_async_to_lds_b128)
#    define HAVE_ASYNC 1
#  endif
#endif
#ifndef HAVE_ASYNC
#  define HAVE_ASYNC 0
#endif

#if HAVE_ASYNC
#  if __has_builtin(__builtin_amdgcn_s_wait_asynccnt)
#    define WAIT_ASYNC(n) __builtin_amdgcn_s_wait_asynccnt(n)
#  else
#    define WAIT_ASYNC(n) asm volatile("s_wait_asynccnt %0" ::"i"(n) : "memory")
#  endif

typedef int __attribute__((ext_vector_type(4))) i4;
// Copy one 16x64 f32 tile (4KB, contiguous) global -> LDS with 8 async b128
// issues (each lane moves 16B per issue).  Tracked by ASYNCcnt, in-order.
static __device__ __forceinline__ void async_copy_tile4k(const float* gsrc,
                                                         float* ldst,
                                                         int lane) {
  __attribute__((address_space(1))) i4* g =
      (__attribute__((address_space(1))) i4*)(uintptr_t)gsrc + lane;
  __attribute__((address_space(3))) i4* l =
      (__attribute__((address_space(3))) i4*)(uint32_t)(uintptr_t)ldst + lane;
#pragma unroll
  for (int c = 0; c < 8; ++c)
    __builtin_amdgcn_global_load_async_to_lds_b128(g + c * 32, l + c * 32, 0, 0);
}
#endif

// D = A(16x4,f32) * B(4x16,f32) + C(16x16,f32), one wave, wave32.
static __device__ __forceinline__ v8f wmma4(v2f a, v2f b, v8f c) {
  return __builtin_amdgcn_wmma_f32_16x16x4_f32(
      /*neg_a=*/false, a, /*neg_b=*/false, b,
      /*c_mod=*/(short)0, c, /*reuse_a=*/false, /*reuse_b=*/false);
}

// ---------------------------------------------------------------------------
// Kernel 1: qkv = x @ Wqkv_w.T + Wqkv_b ; scatter to Q / (K*scale) / V in
// [B, H, S, 64] layout.  One wave computes a 32x64 tile of the 4096x3072
// output (2 row-tiles x 4 col-tiles, 8 WMMA per K-step of 4).
// ---------------------------------------------------------------------------
__global__ void qkv_gemm(const float* __restrict__ x,
                         const float* __restrict__ w,     // [3072,1024]
                         const float* __restrict__ bias,  // [3072]
                         float* __restrict__ q,
                         float* __restrict__ k,
                         float* __restrict__ v) {
  const int colBlocks = (3 * D_MODEL) / 64;                // 48
  int wave = blockIdx.x * (blockDim.x >> 5) + (threadIdx.x >> 5);
  int rt = wave / colBlocks;                               // 32-row block
  int cb = wave % colBlocks;                               // 64-col block
  int lane = threadIdx.x & 31;
  int r = lane & 15;
  int h = lane >> 4;

  const float* a0 = x + (size_t)(rt * 32 + r) * D_MODEL;
  const float* a1 = a0 + (size_t)16 * D_MODEL;
  const float* b0 = w + (size_t)(cb * 64 +  0 + r) * D_MODEL;
  const float* b1 = w + (size_t)(cb * 64 + 16 + r) * D_MODEL;
  const float* b2 = w + (size_t)(cb * 64 + 32 + r) * D_MODEL;
  const float* b3 = w + (size_t)(cb * 64 + 48 + r) * D_MODEL;

  v8f acc[2][4] = {};
  for (int kb0 = 0; kb0 < D_MODEL; kb0 += 32) {
    if (kb0 + 32 < D_MODEL) {            // global_prefetch_b8 of next K-slab
      __builtin_prefetch(a0 + kb0 + 32, 0, 1);
      __builtin_prefetch(a1 + kb0 + 32, 0, 1);
      __builtin_prefetch(b0 + kb0 + 32, 0, 1);
      __builtin_prefetch(b1 + kb0 + 32, 0, 1);
      __builtin_prefetch(b2 + kb0 + 32, 0, 1);
      __builtin_prefetch(b3 + kb0 + 32, 0, 1);
    }
#pragma unroll
    for (int kk = 0; kk < 32; kk += 4) {
      int ko = kb0 + kk + 2 * h;
      v2f af0 = *(const v2f*)(a0 + ko);
      v2f af1 = *(const v2f*)(a1 + ko);
      v2f bf0 = *(const v2f*)(b0 + ko);
      v2f bf1 = *(const v2f*)(b1 + ko);
      v2f bf2 = *(const v2f*)(b2 + ko);
      v2f bf3 = *(const v2f*)(b3 + ko);
      acc[0][0] = wmma4(af0, bf0, acc[0][0]);
      acc[0][1] = wmma4(af0, bf1, acc[0][1]);
      acc[0][2] = wmma4(af0, bf2, acc[0][2]);
      acc[0][3] = wmma4(af0, bf3, acc[0][3]);
      acc[1][0] = wmma4(af1, bf0, acc[1][0]);
      acc[1][1] = wmma4(af1, bf1, acc[1][1]);
      acc[1][2] = wmma4(af1, bf2, acc[1][2]);
      acc[1][3] = wmma4(af1, bf3, acc[1][3]);
    }
  }

  int col0   = cb * 64;                // 64-aligned -> one region, one head
  int region = col0 >> 10;             // 0=q 1=k 2=v
  int head   = (col0 & 1023) >> 6;
  float sc   = (region == 1) ? KSCALE : 1.0f;
  float* dst = (region == 0) ? q : (region == 1) ? k : v;
  float bv[4];
#pragma unroll
  for (int nb = 0; nb < 4; ++nb) bv[nb] = bias[col0 + nb * 16 + r];

#pragma unroll
  for (int u = 0; u < 2; ++u) {
#pragma unroll
    for (int nb = 0; nb < 4; ++nb) {
#pragma unroll
      for (int j = 0; j < 8; ++j) {
        int t = rt * 32 + u * 16 + j + 8 * h;
        int b = t >> 11;
        int s = t & (SEQ - 1);
        dst[((size_t)(b * NUM_HEADS + head) * SEQ + s) * HEAD_DIM +
            nb * 16 + r] = (acc[u][nb][j] + bv[nb]) * sc;
      }
    }
  }
}

// ---------------------------------------------------------------------------
// Kernel 2: causal flash attention, f32 WMMA.  One wave = 32 query rows of
// one (b,h): each K/V fragment feeds two WMMAs.  K is pre-scaled.
// V tiles are double-buffered into LDS via async copies when available.
// Q,K,V,O all in [B,H,S,64] layout.
// ---------------------------------------------------------------------------
__global__ void attn(const float* __restrict__ Q,
                     const float* __restrict__ K,
                     const float* __restrict__ V,
                     float* __restrict__ O) {
  __shared__ float pbuf[4][2 * 256];                       // per-wave P tiles
#if HAVE_ASYNC
  __shared__ float vstage[4][2][16 * HEAD_DIM];            // per-wave V tiles
#endif
  int wslot = threadIdx.x >> 5;
  int wave  = blockIdx.x * (blockDim.x >> 5) + wslot;
  const int qBlocks = SEQ / 32;                            // 64
  int bh = wave / qBlocks;
  int qt = wave % qBlocks;                                 // 32-row q block
  int lane = threadIdx.x & 31;
  int r = lane & 15;
  int h = lane >> 4;

  const float* qb  = Q + (size_t)bh * SEQ * HEAD_DIM;
  const float* kbp = K + (size_t)bh * SEQ * HEAD_DIM;
  const float* vbp = V + (size_t)bh * SEQ * HEAD_DIM;

  // Preload the 32x64 Q block as 2x16 A-fragments (reused every KV tile).
  v2f qa[2][16];
  {
    const float* q0 = qb + (size_t)(qt * 32 + r) * HEAD_DIM;
    const float* q1 = q0 + (size_t)16 * HEAD_DIM;
#pragma unroll
    for (int c = 0; c < 16; ++c) {
      qa[0][c] = *(const v2f*)(q0 + c * 4 + 2 * h);
      qa[1][c] = *(const v2f*)(q1 + c * 4 + 2 * h);
    }
  }

  v8f acc[2][4] = {};
  float m[2][8], l[2][8];
#pragma unroll
  for (int u = 0; u < 2; ++u)
#pragma unroll
    for (int j = 0; j < 8; ++j) { m[u][j] = -1e30f; l[u][j] = 0.0f; }

  const int kjmax = 2 * qt + 1;
#if HAVE_ASYNC
  async_copy_tile4k(vbp, vstage[wslot][0], lane);          // prime tile 0
#endif
  for (int kj = 0; kj <= kjmax; ++kj) {
    const float* krow = kbp + (size_t)(kj * 16 + r) * HEAD_DIM;
    const float* vrow = vbp + (size_t)(kj * 16) * HEAD_DIM;
    if (kj < kjmax) {                    // prefetch next KV tile
      __builtin_prefetch(krow + 16 * HEAD_DIM, 0, 1);
      __builtin_prefetch(vrow + 16 * HEAD_DIM, 0, 1);
    }
#if HAVE_ASYNC
    // stream next V tile while computing on the current one
    if (kj < kjmax) {
      async_copy_tile4k(vbp + (size_t)((kj + 1) * 16) * HEAD_DIM,
                        vstage[wslot][(kj + 1) & 1], lane);
      WAIT_ASYNC(8);                     // 8 newest (next tile) may remain
    } else {
      WAIT_ASYNC(0);
    }
    __builtin_amdgcn_wave_barrier();
    const float* vl = vstage[wslot][kj & 1];
#endif
    // ---- scores S = Q * K^T (2x16 WMMAs, K=64; K-frag reused twice) ----
    v8f s[2] = {};
#pragma unroll
    for (int c = 0; c < 16; ++c) {
      v2f bf = *(const v2f*)(krow + c * 4 + 2 * h);
      s[0] = wmma4(qa[0][c], bf, s[0]);
      s[1] = wmma4(qa[1][c], bf, s[1]);
    }
    // causal mask (covers diagonal tiles and the fully-masked u=0/kj=max case)
    if (kj >= 2 * qt) {
#pragma unroll
      for (int u = 0; u < 2; ++u)
#pragma unroll
        for (int j = 0; j < 8; ++j) {
          int grow = (2 * qt + u) * 16 + j + 8 * h;
          if (kj * 16 + r > grow) s[u][j] += NEGMASK;
        }
    }
    // ---- online softmax (rows j+8h in lanes sharing h; cols across r) ----
#pragma unroll
    for (int u = 0; u < 2; ++u) {
      float p[8];
#pragma unroll
      for (int j = 0; j < 8; ++j) {
        float rm = s[u][j];
        rm = fmaxf(rm, __shfl_xor(rm, 1, 32));
        rm = fmaxf(rm, __shfl_xor(rm, 2, 32));
        rm = fmaxf(rm, __shfl_xor(rm, 4, 32));
        rm = fmaxf(rm, __shfl_xor(rm, 8, 32));
        float mn    = fmaxf(m[u][j], rm);
        float alpha = __expf(m[u][j] - mn);
        float pv    = __expf(s[u][j] - mn);
        float rs = pv;
        rs += __shfl_xor(rs, 1, 32);
        rs += __shfl_xor(rs, 2, 32);
        rs += __shfl_xor(rs, 4, 32);
        rs += __shfl_xor(rs, 8, 32);
        l[u][j] = l[u][j] * alpha + rs;
        m[u][j] = mn;
        acc[u][0][j] *= alpha; acc[u][1][j] *= alpha;
        acc[u][2][j] *= alpha; acc[u][3][j] *= alpha;
        p[j] = pv;
      }
      float* pl = pbuf[wslot] + u * 256;
#pragma unroll
      for (int j = 0; j < 8; ++j) pl[(j + 8 * h) * 16 + r] = p[j];
    }
    __builtin_amdgcn_wave_barrier();     // keep DS store->load ordered in wave
    v2f pa[2][4];
#pragma unroll
    for (int u = 0; u < 2; ++u) {
      const float* pl = pbuf[wslot] + u * 256;
#pragma unroll
      for (int c = 0; c < 4; ++c)
        pa[u][c] = *(const v2f*)(pl + r * 16 + c * 4 + 2 * h);
    }
    __builtin_amdgcn_wave_barrier();
    // ---- O += P * V (4 K-steps x 4 N-tiles x 2 row-tiles = 32 WMMAs) ----
#pragma unroll
    for (int c = 0; c < 4; ++c) {
      int k0 = c * 4 + 2 * h;
#pragma unroll
      for (int nb = 0; nb < 4; ++nb) {
        v2f bf;
#if HAVE_ASYNC
        bf.x = vl[(k0    ) * HEAD_DIM + nb * 16 + r];
        bf.y = vl[(k0 + 1) * HEAD_DIM + nb * 16 + r];
#else
        bf.x = vrow[(k0    ) * HEAD_DIM + nb * 16 + r];
        bf.y = vrow[(k0 + 1) * HEAD_DIM + nb * 16 + r];
#endif
        acc[0][nb] = wmma4(pa[0][c], bf, acc[0][nb]);
        acc[1][nb] = wmma4(pa[1][c], bf, acc[1][nb]);
      }
    }
  }

  float* ob = O + (size_t)(bh * SEQ + qt * 32) * HEAD_DIM;
#pragma unroll
  for (int u = 0; u < 2; ++u)
#pragma unroll
    for (int j = 0; j < 8; ++j) {
      float inv = 1.0f / l[u][j];
      float* orow = ob + (size_t)(u * 16 + j + 8 * h) * HEAD_DIM;
      orow[ 0 + r] = acc[u][0][j] * inv;
      orow[16 + r] = acc[u][1][j] * inv;
      orow[32 + r] = acc[u][2][j] * inv;
      orow[48 + r] = acc[u][3][j] * inv;
    }
}

// ---------------------------------------------------------------------------
// Kernel 3: out = ctx.reshape(B,S,D) @ out_w.T + out_b
// ctx is in [B,H,S,64] layout -> logical k = head*64+dim.  32x64 per wave.
// ---------------------------------------------------------------------------
__global__ void out_gemm(const float* __restrict__ ctx,
                         const float* __restrict__ w,     // [1024,1024]
                         const float* __restrict__ bias,  // [1024]
                         float* __restrict__ out) {
  const int colBlocks = D_MODEL / 64;                      // 16
  int wave = blockIdx.x * (blockDim.x >> 5) + (threadIdx.x >> 5);
  int rt = wave / colBlocks;
  int cb = wave % colBlocks;
  int lane = threadIdx.x & 31;
  int r = lane & 15;
  int h = lane >> 4;

  int t0 = rt * 32 + r;                 // 32-row block stays in one batch
  int bb = t0 >> 11;
  int s0 = t0 & (SEQ - 1);
  const float* b0 = w + (size_t)(cb * 64 +  0 + r) * D_MODEL;
  const float* b1 = w + (size_t)(cb * 64 + 16 + r) * D_MODEL;
  const float* b2 = w + (size_t)(cb * 64 + 32 + r) * D_MODEL;
  const float* b3 = w + (size_t)(cb * 64 + 48 + r) * D_MODEL;

  v8f acc[2][4] = {};
  for (int head = 0; head < NUM_HEADS; ++head) {
    const float* a0 =
        ctx + ((size_t)(bb * NUM_HEADS + head) * SEQ + s0) * HEAD_DIM;
    const float* a1 = a0 + (size_t)16 * HEAD_DIM;
    if (head + 1 < NUM_HEADS) {
      __builtin_prefetch(a0 + (size_t)SEQ * HEAD_DIM, 0, 1);
      __builtin_prefetch(b0 + (head + 1) * 64, 0, 1);
    }
#pragma unroll
    for (int kk = 0; kk < 64; kk += 4) {
      int ko = kk + 2 * h;
      int kw = head * 64 + ko;
      v2f af0 = *(const v2f*)(a0 + ko);
      v2f af1 = *(const v2f*)(a1 + ko);
      v2f bf0 = *(const v2f*)(b0 + kw);
      v2f bf1 = *(const v2f*)(b1 + kw);
      v2f bf2 = *(const v2f*)(b2 + kw);
      v2f bf3 = *(const v2f*)(b3 + kw);
      acc[0][0] = wmma4(af0, bf0, acc[0][0]);
      acc[0][1] = wmma4(af0, bf1, acc[0][1]);
      acc[0][2] = wmma4(af0, bf2, acc[0][2]);
      acc[0][3] = wmma4(af0, bf3, acc[0][3]);
      acc[1][0] = wmma4(af1, bf0, acc[1][0]);
      acc[1][1] = wmma4(af1, bf1, acc[1][1]);
      acc[1][2] = wmma4(af1, bf2, acc[1][2]);
      acc[1][3] = wmma4(af1, bf3, acc[1][3]);
    }
  }

  float bv[4];
#pragma unroll
  for (int nb = 0; nb < 4; ++nb) bv[nb] = bias[cb * 64 + nb * 16 + r];
#pragma unroll
  for (int u = 0; u < 2; ++u)
#pragma unroll
    for (int nb = 0; nb < 4; ++nb)
#pragma unroll
      for (int j = 0; j < 8; ++j) {
        int t = rt * 32 + u * 16 + j + 8 * h;
        out[(size_t)t * D_MODEL + cb * 64 + nb * 16 + r] =
            acc[u][nb][j] + bv[nb];
      }
}

// ---------------------------------------------------------------------------
extern "C" void kernel_launch(void* const* d_in, const int* in_sizes, int n_in,
                              void* d_out, int out_size, void* d_ws,
                              size_t ws_size, hipStream_t stream) {
  const float* x      = (const float*)d_in[0];
  const float* Wqkv_w = (const float*)d_in[1];
  const float* Wqkv_b = (const float*)d_in[2];
  const float* out_w  = (const float*)d_in[3];
  const float* out_b  = (const float*)d_in[4];
  float* out = (float*)d_out;

  const size_t per = (size_t)BATCH * NUM_HEADS * SEQ * HEAD_DIM;  // 4M floats
  float* q    = (float*)d_ws;
  float* k    = q + per;
  float* v    = k + per;
  float* ctxA = v + per;
  float* ctxB = ctxA + per;   // total 5*16MB = 80MB workspace

  // 1) QKV projection: 128 x 48 blocks of 32x64, 4 waves/block
  {
    int waves = (BATCH * SEQ / 32) * (3 * D_MODEL / 64);   // 6144
    qkv_gemm<<<waves / 4, 128, 0, stream>>>(x, Wqkv_w, Wqkv_b, q, k, v);
  }
  // 2) 3x causal attention (K pre-scaled), ping-pong contexts
  {
    int waves = BATCH * NUM_HEADS * (SEQ / 32);            // 2048
    attn<<<waves / 4, 128, 0, stream>>>(q,    k, v, ctxA);
    attn<<<waves / 4, 128, 0, stream>>>(ctxA, k, v, ctxB);
    attn<<<waves / 4, 128, 0, stream>>>(ctxB, k, v, ctxA);
  }
  // 3) Output projection: 128 x 16 blocks of 32x64
  {
    int waves = (BATCH * SEQ / 32) * (D_MODEL / 64);       // 2048
    out_gemm<<<waves / 4, 128, 0, stream>>>(ctxA, out_w, out_b, out);
  }
}